// Model_89704686944629
// MI455X (gfx1250) — compile-verified
//
#include <hip/hip_runtime.h>
#include <math.h>

// ---------------------------------------------------------------------------
// Ray-triangle (Moller-Trumbore) reformulated as an fp32 WMMA GEMM.
//
//   a      = dir . n                 n   = e2 x e1
//   u_num  = m . e2 + dir . (v0xe2)  m   = origin x dir
//   v_num  = -m . e1 - dir . (v0xe1)
//   t_num  = -origin . n + (v0 . n)
//   f = 1/(a+eps); u=f*u_num; v=f*v_num; t=f*t_num; validity + min over tris.
//
// K (=12, padded) order: 0..2 dir, 3..5 m, 6..8 origin, 9 = 1, 10..11 = 0.
// K-pairs are interleaved innermost so every WMMA fragment (v2f holding
// K=2P,2P+1) is one aligned global_load_b64:
//   F2[pair][ray][2]      (ray features, pair-major)
//   G2[pair][o][tri][2]   (triangle features, o = 0:a 1:u_num 2:v_num 3:t_num)
// All 12 B loads per tile index off ONE per-lane base pointer with
// compile-time-constant offsets (< 2^23 B) so they fold into the IOFFSET
// field of global_load_b64 — one address add per tile.
// Pair GEMM done with chained V_WMMA_F32_16X16X4_F32 (3 per output tile).
// ---------------------------------------------------------------------------

typedef __attribute__((ext_vector_type(2))) float v2f;
typedef __attribute__((ext_vector_type(8))) float v8f;

#define R_RAYS 8192
#define T_TRIS 2048
#define NPAIR  6            // K=12 as 6 aligned pairs
#define EPSF   1e-7f
#define TSPLIT 4            // triangle-range split across gridDim.y
#define WPB    8            // waves per block (256 threads, wave32)

__device__ __forceinline__ void cross3(float ax, float ay, float az,
                                       float bx, float by, float bz,
                                       float& cx, float& cy, float& cz) {
  cx = ay * bz - az * by;
  cy = az * bx - ax * bz;
  cz = ax * by - ay * bx;
}

__device__ __forceinline__ v2f mk2(float x, float y) {
  v2f r; r.x = x; r.y = y; return r;
}

// Kernel 1: build F2 (ray features), G2 (triangle features), init out = +inf.
__global__ void __launch_bounds__(256)
features_kernel(const float* __restrict__ ro, const float* __restrict__ rd,
                const float* __restrict__ p0, const float* __restrict__ p1,
                const float* __restrict__ p2,
                float* __restrict__ F2, float* __restrict__ G2,
                unsigned int* __restrict__ out_bits) {
  const int i = blockIdx.x * blockDim.x + threadIdx.x;

  if (i < R_RAYS) {
    const float ox = ro[3 * i], oy = ro[3 * i + 1], oz = ro[3 * i + 2];
    const float dx = rd[3 * i], dy = rd[3 * i + 1], dz = rd[3 * i + 2];
    float mx, my, mz;
    cross3(ox, oy, oz, dx, dy, dz, mx, my, mz);   // m = origin x dir
    v2f* F = (v2f*)F2;
    F[0 * R_RAYS + i] = mk2(dx, dy);
    F[1 * R_RAYS + i] = mk2(dz, mx);
    F[2 * R_RAYS + i] = mk2(my, mz);
    F[3 * R_RAYS + i] = mk2(ox, oy);
    F[4 * R_RAYS + i] = mk2(oz, 1.0f);
    F[5 * R_RAYS + i] = mk2(0.0f, 0.0f);
    out_bits[i] = 0x7F800000u;                    // +inf (all mins positive)
  }

  if (i < T_TRIS) {
    const float ax = p0[3 * i], ay = p0[3 * i + 1], az = p0[3 * i + 2];
    const float bx = p1[3 * i], by = p1[3 * i + 1], bz = p1[3 * i + 2];
    const float cx = p2[3 * i], cy = p2[3 * i + 1], cz = p2[3 * i + 2];
    const float e1x = bx - ax, e1y = by - ay, e1z = bz - az;
    const float e2x = cx - ax, e2y = cy - ay, e2z = cz - az;
    float nx, ny, nz;                             // n = e2 x e1
    cross3(e2x, e2y, e2z, e1x, e1y, e1z, nx, ny, nz);
    float c2x, c2y, c2z;                          // v0 x e2
    cross3(ax, ay, az, e2x, e2y, e2z, c2x, c2y, c2z);
    float c1x, c1y, c1z;                          // v0 x e1
    cross3(ax, ay, az, e1x, e1y, e1z, c1x, c1y, c1z);
    const float d = ax * nx + ay * ny + az * nz;  // v0 . n

    v2f* G = (v2f*)G2;
    const v2f z2 = mk2(0.0f, 0.0f);
    // index helper: pair P, output o, triangle i
#define GIDX(P, o) (((P) * 4 + (o)) * T_TRIS + i)
    // o=0 : a = dir.n                  (K 0..2 = n)
    G[GIDX(0, 0)] = mk2(nx, ny);
    G[GIDX(1, 0)] = mk2(nz, 0.0f);
    G[GIDX(2, 0)] = z2;
    G[GIDX(3, 0)] = z2;
    G[GIDX(4, 0)] = z2;
    G[GIDX(5, 0)] = z2;
    // o=1 : u_num = dir.(v0xe2) + m.e2 (K 0..2 = v0xe2, K 3..5 = e2)
    G[GIDX(0, 1)] = mk2(c2x, c2y);
    G[GIDX(1, 1)] = mk2(c2z, e2x);
    G[GIDX(2, 1)] = mk2(e2y, e2z);
    G[GIDX(3, 1)] = z2;
    G[GIDX(4, 1)] = z2;
    G[GIDX(5, 1)] = z2;
    // o=2 : v_num = -dir.(v0xe1) - m.e1
    G[GIDX(0, 2)] = mk2(-c1x, -c1y);
    G[GIDX(1, 2)] = mk2(-c1z, -e1x);
    G[GIDX(2, 2)] = mk2(-e1y, -e1z);
    G[GIDX(3, 2)] = z2;
    G[GIDX(4, 2)] = z2;
    G[GIDX(5, 2)] = z2;
    // o=3 : t_num = -origin.n + 1*(v0.n)  (K 6..8 = -n, K 9 = v0.n)
    G[GIDX(0, 3)] = z2;
    G[GIDX(1, 3)] = z2;
    G[GIDX(2, 3)] = z2;
    G[GIDX(3, 3)] = mk2(-nx, -ny);
    G[GIDX(4, 3)] = mk2(-nz, d);
    G[GIDX(5, 3)] = z2;
#undef GIDX
  }
}

// Kernel 2: each wave32 owns a 16-ray strip; loops over 16-triangle tiles.
// Per tile: 4 output D tiles (a, u_num, v_num, t_num), each K=12 via 3
// chained V_WMMA_F32_16X16X4_F32. Epilogue is packed-f32 VALU and
// co-executes with the XDL WMMA pipe.
__global__ void __launch_bounds__(256)
raytri_wmma_kernel(const float* __restrict__ F2, const float* __restrict__ G2,
                   unsigned int* __restrict__ out_bits) {
  const int lane = threadIdx.x & 31;
  const int wave = threadIdx.x >> 5;
  const int m    = lane & 15;    // A: M row / B,D: N col (triangle)
  const int hi   = lane >> 4;    // K-half select within a 4-chunk

  const int ray_base = (blockIdx.x * WPB + wave) * 16;
  const int t_begin  = blockIdx.y * (T_TRIS / TSPLIT);
  const int t_end    = t_begin + (T_TRIS / TSPLIT);

  // A fragments (16x4 f32): lane 0-15 -> K={0,1}; lane 16-31 -> K={2,3} of
  // each chunk; chunk c uses pair P = 2c + hi. hi folded into the base so
  // offsets are compile-time constants. Loaded once, reused for all tiles.
  const v2f* fp = (const v2f*)F2 + (size_t)hi * R_RAYS + ray_base + m;
  const v2f a0 = fp[0 * R_RAYS];
  const v2f a1 = fp[2 * R_RAYS];
  const v2f a2 = fp[4 * R_RAYS];

  float minT[8];
#pragma unroll
  for (int j = 0; j < 8; ++j) minT[j] = __builtin_inff();

  // Single per-lane B base pointer; advances 16 v2f (128 B) per tile. All
  // 12 fragment offsets (8c+o)*T_TRIS are constants < 2^23 B -> IOFFSET.
  const v2f* gp = (const v2f*)G2 + (size_t)hi * 4 * T_TRIS + t_begin + m;

  for (int tb = t_begin; tb < t_end; tb += 16, gp += 16) {
    v8f acc[4];
#pragma unroll
    for (int o = 0; o < 4; ++o) {
      // B fragments (4x16 f32): lane&15 = N (triangle), K pair = 2c + hi.
      const v2f b0 = gp[(0 * 8 + o) * T_TRIS];
      const v2f b1 = gp[(1 * 8 + o) * T_TRIS];
      const v2f b2 = gp[(2 * 8 + o) * T_TRIS];

      v8f c = {};
      c = __builtin_amdgcn_wmma_f32_16x16x4_f32(false, a0, false, b0,
                                                (short)0, c, false, false);
      c = __builtin_amdgcn_wmma_f32_16x16x4_f32(false, a1, false, b1,
                                                (short)0, c, false, false);
      c = __builtin_amdgcn_wmma_f32_16x16x4_f32(false, a2, false, b2,
                                                (short)0, c, false, false);
      acc[o] = c;
    }

    // Epilogue: D layout -> VGPR j = ray (j + 8*hi strip), lane = triangle.
#pragma unroll
    for (int j = 0; j < 8; ++j) {
      const float av = acc[0][j];
      const float un = acc[1][j];
      const float vn = acc[2][j];
      const float tn = acc[3][j];
      const float f  = __builtin_amdgcn_rcpf(av + EPSF);
      const float u  = f * un;
      const float v  = f * vn;
      const float t  = f * tn;
      const bool valid = (__builtin_fabsf(av) >= EPSF) & (u >= 0.0f) &
                         (v >= 0.0f) & ((u + v) <= 1.0f) & (t > EPSF);
      const float th = valid ? t : __builtin_inff();
      minT[j] = fminf(minT[j], th);
    }
  }

  // Min-reduce across the 16 lanes of each half (xor masks < 16 stay in-half).
#pragma unroll
  for (int j = 0; j < 8; ++j) {
#pragma unroll
    for (int off = 1; off < 16; off <<= 1)
      minT[j] = fminf(minT[j], __shfl_xor(minT[j], off, 32));
  }

  // Lanes 0 and 16 publish rays [ray_base..+7] and [ray_base+8..+15].
  // Candidates are positive (or +inf) -> float order == uint bit order.
  if (m == 0) {
    const int rb = ray_base + hi * 8;
#pragma unroll
    for (int j = 0; j < 8; ++j)
      atomicMin(&out_bits[rb + j], __float_as_uint(minT[j]));
  }
}

extern "C" void kernel_launch(void* const* d_in, const int* in_sizes, int n_in,
                              void* d_out, int out_size, void* d_ws,
                              size_t ws_size, hipStream_t stream) {
  const float* ro = (const float*)d_in[0];
  const float* rd = (const float*)d_in[1];
  const float* v0 = (const float*)d_in[2];
  const float* v1 = (const float*)d_in[3];
  const float* v2 = (const float*)d_in[4];

  // Workspace: F2 = 6*8192*2 floats (384 KB), G2 = 6*4*2048*2 floats (384 KB).
  float* F2 = (float*)d_ws;
  float* G2 = F2 + (size_t)NPAIR * R_RAYS * 2;
  unsigned int* out_bits = (unsigned int*)d_out;

  hipLaunchKernelGGL(features_kernel, dim3((R_RAYS + 255) / 256), dim3(256), 0,
                     stream, ro, rd, v0, v1, v2, F2, G2, out_bits);

  dim3 grid(R_RAYS / (16 * WPB), TSPLIT);   // 64 x 4 blocks, 2048 waves
  hipLaunchKernelGGL(raytri_wmma_kernel, grid, dim3(256), 0, stream, F2, G2,
                     out_bits);
}